// SparseEmbedding_29609504539057
// MI455X (gfx1250) — compile-verified
//
#include <hip/hip_runtime.h>

typedef __attribute__((ext_vector_type(2))) float v2f;
typedef __attribute__((ext_vector_type(8))) float v8f;

#define DIST_SCALE_INV 0.1f
#define CUTOFF_INV     0.2f

__device__ __forceinline__ float fast_silu(float x) {
  // silu(x) = x * sigmoid(x) = x / (1 + exp(-x));  exp via v_exp_f32 (2^x)
  float e = __builtin_amdgcn_exp2f(-1.44269504088896340736f * x);
  return x * __builtin_amdgcn_rcpf(1.0f + e);
}

__device__ __forceinline__ v8f wmma4(v2f a, v2f b, v8f c) {
  // V_WMMA_F32_16X16X4_F32 : D = A(16x4) * B(4x16) + C(16x16), full fp32
  return __builtin_amdgcn_wmma_f32_16x16x4_f32(false, a, false, b, (short)0, c,
                                               false, false);
}

// ---------------------------------------------------------------------------
// Kernel 1: h1 = MLP1(r/10)   [16384,3] -> 256 -> 256 -> 256  (silu, silu, -)
// Block: 256 threads (8 waves), one 16-row tile; wave w owns cols [32w,32w+32).
// ---------------------------------------------------------------------------
__global__ __launch_bounds__(256) void h1_kernel(
    const float* __restrict__ r,
    const float* __restrict__ W0, const float* __restrict__ b0,
    const float* __restrict__ W1, const float* __restrict__ b1,
    const float* __restrict__ W2, const float* __restrict__ b2,
    float* __restrict__ out)
{
  __shared__ float actA[16 * 260];   // pad-4 stride: conflict-free A loads
  __shared__ float actB[16 * 260];

  const int t    = threadIdx.x;
  const int lane = t & 31;
  const int wave = t >> 5;
  const int ln   = lane & 15;        // A layout: row (M) = ln
  const int hl   = lane >> 4;        // half-wave selects K pair / row+8
  const long rbase = (long)blockIdx.x * 16;

  // ---- layer 0: [16,3]@[3,256]+b0, silu. thread t owns column t ----
  {
    const float w0x = W0[t], w0y = W0[256 + t], w0z = W0[512 + t], bb = b0[t];
    for (int row = 0; row < 16; ++row) {
      const float* rp = r + (rbase + row) * 3;
      float x = fmaf(rp[0] * DIST_SCALE_INV, w0x,
                fmaf(rp[1] * DIST_SCALE_INV, w0y,
                fmaf(rp[2] * DIST_SCALE_INV, w0z, bb)));
      actA[row * 260 + t] = fast_silu(x);
    }
  }
  __syncthreads();

  const int c0 = (wave * 2) * 16 + ln;   // ntile 0 column for this lane
  const int c1 = c0 + 16;                // ntile 1 column

  // ---- layer 1: [16,256]@[256,256]+b1, silu ----
  {
    v8f acc0 = {}, acc1 = {};
#pragma unroll 8
    for (int k = 0; k < 64; ++k) {
      v2f A = *(const v2f*)&actA[ln * 260 + 4 * k + 2 * hl];
      const float* wp0 = W1 + (4 * k + 2 * hl) * 256 + c0;
      const float* wp1 = W1 + (4 * k + 2 * hl) * 256 + c1;
      v2f B0; B0.x = wp0[0]; B0.y = wp0[256];
      v2f B1v; B1v.x = wp1[0]; B1v.y = wp1[256];
      acc0 = wmma4(A, B0, acc0);
      acc1 = wmma4(A, B1v, acc1);
    }
    const float bv0 = b1[c0], bv1 = b1[c1];
#pragma unroll
    for (int v = 0; v < 8; ++v) {
      actB[(v + 8 * hl) * 260 + c0] = fast_silu(acc0[v] + bv0);
      actB[(v + 8 * hl) * 260 + c1] = fast_silu(acc1[v] + bv1);
    }
  }
  __syncthreads();

  // ---- layer 2: [16,256]@[256,256]+b2 (no activation) -> out ----
  {
    v8f acc0 = {}, acc1 = {};
#pragma unroll 8
    for (int k = 0; k < 64; ++k) {
      v2f A = *(const v2f*)&actB[ln * 260 + 4 * k + 2 * hl];
      const float* wp0 = W2 + (4 * k + 2 * hl) * 256 + c0;
      const float* wp1 = W2 + (4 * k + 2 * hl) * 256 + c1;
      v2f B0; B0.x = wp0[0]; B0.y = wp0[256];
      v2f B1v; B1v.x = wp1[0]; B1v.y = wp1[256];
      acc0 = wmma4(A, B0, acc0);
      acc1 = wmma4(A, B1v, acc1);
    }
    const float bv0 = b2[c0], bv1 = b2[c1];
#pragma unroll
    for (int v = 0; v < 8; ++v) {
      out[(rbase + v + 8 * hl) * 256 + c0] = acc0[v] + bv0;
      out[(rbase + v + 8 * hl) * 256 + c1] = acc1[v] + bv1;
    }
  }
}

// ---------------------------------------------------------------------------
// Kernel 2: h2 = MLP2(diff/10) (silu x3), agg = sum_m kern*h2, out += agg@Wout
// Block: 256 threads, 16 particles (1024 edges). Wave w owns particles 2w,2w+1.
// ---------------------------------------------------------------------------
__global__ __launch_bounds__(256) void h2_kernel(
    const float* __restrict__ diff, const float* __restrict__ dist,
    const float* __restrict__ wgt,
    const float* __restrict__ W0, const float* __restrict__ b0,
    const float* __restrict__ W1, const float* __restrict__ b1,
    const float* __restrict__ W2, const float* __restrict__ b2,
    const float* __restrict__ Wout,
    float* __restrict__ out)
{
  __shared__ float s_kern[1024];          // envelope*weight per edge
  __shared__ float s_W1[1024];            // swizzled B-pair layout
  __shared__ float s_W2[1024];
  __shared__ float s_W0[96];
  __shared__ float s_b0[32], s_b1[32], s_b2[32];
  __shared__ float s_agg[16 * 34];        // agg[particle][feat], pad stride 34
  __shared__ float s_bounce[8 * 16 * 34]; // per-wave transpose tile

  const int t    = threadIdx.x;
  const int lane = t & 31;
  const int wave = t >> 5;
  const int ln   = lane & 15;
  const int hl   = lane >> 4;
  const long pbase = (long)blockIdx.x * 16;   // first particle row
  const long ebase = pbase * 64;              // first edge

  // ---- stage envelope kernel: cut(d)=1-15d^4+24d^5-10d^6, d=dist/5 ----
  for (int i = t; i < 1024; i += 256) {
    float d  = dist[ebase + i] * CUTOFF_INV;
    float d2 = d * d;
    float d4 = d2 * d2;
    float cut = fmaf(d4, fmaf(d, fmaf(d, -10.0f, 24.0f), -15.0f), 1.0f);
    cut = (d < 1.0f) ? cut : 0.0f;
    s_kern[i] = cut * wgt[ebase + i];
  }
  // ---- stage 32x32 weights in B-pair layout: (W[4k+2hi][c], W[4k+2hi+1][c]) ----
  for (int i = t; i < 1024; i += 256) {
    int kr = i >> 5, c = i & 31;
    int k = kr >> 2, j = kr & 3, hi = j >> 1, lo = j & 1;
    s_W1[k * 128 + c * 4 + hi * 2 + lo] = W1[i];
    s_W2[k * 128 + c * 4 + hi * 2 + lo] = W2[i];
  }
  for (int i = t; i < 96; i += 256) s_W0[i] = W0[i];
  if (t < 32) { s_b0[t] = b0[t]; s_b1[t] = b1[t]; s_b2[t] = b2[t]; }
  __syncthreads();

  // ---- resident B operands for layers 1,2 (64 VGPRs) + bias scalars ----
  v2f B1r[2][8], B2r[2][8];
#pragma unroll
  for (int nt = 0; nt < 2; ++nt)
#pragma unroll
    for (int k = 0; k < 8; ++k) {
      int c = nt * 16 + ln;
      B1r[nt][k] = *(const v2f*)&s_W1[k * 128 + c * 4 + hl * 2];
      B2r[nt][k] = *(const v2f*)&s_W2[k * 128 + c * 4 + hl * 2];
    }
  const float b1v0 = s_b1[ln], b1v1 = s_b1[16 + ln];
  const float b2v0 = s_b2[ln], b2v1 = s_b2[16 + ln];

  float* bounce = &s_bounce[wave * 16 * 34];

  for (int p = 0; p < 2; ++p) {
    const int part = wave * 2 + p;
    float agg0 = 0.0f, agg1 = 0.0f;

    for (int rt = 0; rt < 4; ++rt) {
      const int  tile = part * 4 + rt;          // 16-edge tile within block
      const long e0   = ebase + tile * 16;

      // -- layer 0: scalar FMAs straight into the WMMA A-register layout --
      const float* dp = diff + (e0 + ln) * 3;
      const float dx = dp[0] * DIST_SCALE_INV;
      const float dy = dp[1] * DIST_SCALE_INV;
      const float dz = dp[2] * DIST_SCALE_INV;
      v2f A[8];
#pragma unroll
      for (int k = 0; k < 8; ++k) {
        int c = 4 * k + 2 * hl;
        float x0 = fmaf(dx, s_W0[c],     fmaf(dy, s_W0[32 + c],     fmaf(dz, s_W0[64 + c],     s_b0[c])));
        float x1 = fmaf(dx, s_W0[c + 1], fmaf(dy, s_W0[32 + c + 1], fmaf(dz, s_W0[64 + c + 1], s_b0[c + 1])));
        A[k].x = fast_silu(x0);
        A[k].y = fast_silu(x1);
      }

      // -- layer 1: [16,32]@[32,32] --
      v8f a0 = {}, a1 = {};
#pragma unroll
      for (int k = 0; k < 8; ++k) {
        a0 = wmma4(A[k], B1r[0][k], a0);
        a1 = wmma4(A[k], B1r[1][k], a1);
      }
      // bias+silu, bounce through LDS (C/D layout -> row-major, stride 34)
#pragma unroll
      for (int v = 0; v < 8; ++v) {
        int row = v + 8 * hl;
        bounce[row * 34 + ln]      = fast_silu(a0[v] + b1v0);
        bounce[row * 34 + 16 + ln] = fast_silu(a1[v] + b1v1);
      }
      asm volatile("" ::: "memory");   // per-wave LDS is HW-ordered; fence compiler

      // -- layer 2: reload in A layout, [16,32]@[32,32] --
      v2f A2[8];
#pragma unroll
      for (int k = 0; k < 8; ++k)
        A2[k] = *(const v2f*)&bounce[ln * 34 + 4 * k + 2 * hl];
      asm volatile("" ::: "memory");
      v8f c0v = {}, c1v = {};
#pragma unroll
      for (int k = 0; k < 8; ++k) {
        c0v = wmma4(A2[k], B2r[0][k], c0v);
        c1v = wmma4(A2[k], B2r[1][k], c1v);
      }

      // -- silu + envelope-weighted M-reduction in C/D layout --
      const float* kp = &s_kern[tile * 16 + 8 * hl];
      float kv[8];
#pragma unroll
      for (int v = 0; v < 8; ++v) kv[v] = kp[v];   // broadcast LDS reads
#pragma unroll
      for (int v = 0; v < 8; ++v) {
        agg0 = fmaf(kv[v], fast_silu(c0v[v] + b2v0), agg0);
        agg1 = fmaf(kv[v], fast_silu(c1v[v] + b2v1), agg1);
      }
    }

    // combine half-wave partials (rows 0-7 with rows 8-15)
    agg0 += __shfl_xor(agg0, 16, 32);
    agg1 += __shfl_xor(agg1, 16, 32);
    if (hl == 0) {
      s_agg[part * 34 + ln]      = agg0;
      s_agg[part * 34 + 16 + ln] = agg1;
    }
  }
  __syncthreads();

  // ---- final: out += agg[16,32] @ Wout[32,256]; wave owns 2 col tiles ----
  v2f Aa[8];
#pragma unroll
  for (int k = 0; k < 8; ++k)
    Aa[k] = *(const v2f*)&s_agg[ln * 34 + 4 * k + 2 * hl];
#pragma unroll
  for (int nn = 0; nn < 2; ++nn) {
    const int nt = wave * 2 + nn;
    const int cc = nt * 16 + ln;
    v8f acc = {};
#pragma unroll
    for (int k = 0; k < 8; ++k) {
      const float* wp = Wout + (4 * k + 2 * hl) * 256 + cc;
      v2f Bv; Bv.x = wp[0]; Bv.y = wp[256];
      acc = wmma4(Aa[k], Bv, acc);
    }
#pragma unroll
    for (int v = 0; v < 8; ++v) {
      float* q = out + (pbase + v + 8 * hl) * 256 + cc;
      *q += acc[v];
    }
  }
}

// ---------------------------------------------------------------------------
extern "C" void kernel_launch(void* const* d_in, const int* in_sizes, int n_in,
                              void* d_out, int out_size, void* d_ws, size_t ws_size,
                              hipStream_t stream) {
  (void)out_size; (void)d_ws; (void)ws_size;
  const float* r    = (const float*)d_in[0];   // [16,1024,3]
  const float* diff = (const float*)d_in[1];   // [16,1024,64,3]
  const float* dist = (const float*)d_in[2];   // [16,1024,64]
  const float* wgt  = (const float*)d_in[3];   // [16,1024,64]

  // Classify flattened params by element count (robust to pytree dict order).
  const float* Wout = nullptr;
  const float* m1W[3] = {nullptr, nullptr, nullptr};
  const float* m1b[3] = {nullptr, nullptr, nullptr};
  const float* m2W[3] = {nullptr, nullptr, nullptr};
  const float* m2b[3] = {nullptr, nullptr, nullptr};
  int n65536 = 0, n256 = 0, n1024 = 0, n32 = 0;
  for (int i = 4; i < n_in; ++i) {
    const float* p = (const float*)d_in[i];
    switch (in_sizes[i]) {
      case 8192:  Wout = p; break;                                   // [32,256]
      case 768:   m1W[0] = p; break;                                 // [3,256]
      case 65536: if (n65536 < 2) m1W[1 + n65536++] = p; break;      // [256,256]
      case 256:   if (n256 < 3)   m1b[n256++] = p; break;            // [256]
      case 96:    m2W[0] = p; break;                                 // [3,32]
      case 1024:  if (n1024 < 2)  m2W[1 + n1024++] = p; break;       // [32,32]
      case 32:    if (n32 < 3)    m2b[n32++] = p; break;             // [32]
      default: break;
    }
  }

  // 16384 particle rows -> 1024 blocks of one 16-row tile each.
  h1_kernel<<<dim3(1024), dim3(256), 0, stream>>>(
      r, m1W[0], m1b[0], m1W[1], m1b[1], m1W[2], m1b[2], (float*)d_out);
  h2_kernel<<<dim3(1024), dim3(256), 0, stream>>>(
      diff, dist, wgt, m2W[0], m2b[0], m2W[1], m2b[1], m2W[2], m2b[2], Wout,
      (float*)d_out);
}